// SoftToHardNdEncoder_14267881357913
// MI455X (gfx1250) — compile-verified
//
#include <hip/hip_runtime.h>

typedef float v2f __attribute__((ext_vector_type(2)));
typedef float v4f __attribute__((ext_vector_type(4)));
typedef float v8f __attribute__((ext_vector_type(8)));

#define B_   8
#define C_   128
#define WH_  4096   // W*H
#define L_   16
#define K_   256
#define CD_  8

__device__ __forceinline__ float xor16f(float x) { return __shfl_xor(x, 16, 32); }

// one async 512B (16B/lane x 32 lanes) global->LDS copy; same IOFFSET applies
// to both the LDS destination and the global source per the CDNA5 ISA.
#define ASYNC_CP(OFF)                                                         \
    asm volatile("global_load_async_to_lds_b128 %0, %1, off offset:" #OFF     \
                 :: "v"(ldsoff), "v"(ga) : "memory")

__global__ __launch_bounds__(128, 1)
void vq_soft_hard_kernel(const float* __restrict__ z,
                         const float* __restrict__ codes,
                         float* __restrict__ soft_out,
                         float* __restrict__ hard_out,
                         int*   __restrict__ idx_out)
{
    __shared__ float smem[4 * 2048];     // 8KB codes slice per wave (32KB/block)

    const int lane = threadIdx.x & 31;
    const int wave = threadIdx.x >> 5;   // 0..3
    const int p    = lane & 15;          // position within tile (N index)
    const int h    = lane >> 4;          // half-wave
    const int q0   = blockIdx.x << 4;    // base flattened position (b,w,h)
    const int b    = q0 >> 12;           // 4096 positions per batch image
    const int inB  = q0 & 4095;

    float* const clds = &smem[wave * 2048];
    const unsigned ldsoff = (unsigned)(unsigned long long)(uintptr_t)clds + lane * 16u;

    // gfx1250 prefetch of the next tile's z region (global_prefetch_b8)
    __builtin_prefetch(z + (size_t)b * (C_ * WH_) + inB + 16, 0, 1);

    #pragma unroll 1
    for (int li = 0; li < 4; ++li) {
        const int l = wave + 4 * li;
        const float* zl = z + (size_t)b * (C_ * WH_) + (size_t)(l * CD_) * WH_ + inB + p;
        const float* cl = codes + (size_t)l * (K_ * CD_);

        // ---- stage codes[l] (8KB) into this wave's LDS slice (ASYNCcnt path) ----
        // WAR guard: previous latent's ds reads must have completed.
        asm volatile("s_wait_dscnt 0x0" ::: "memory");
        {
            const unsigned long long ga =
                (unsigned long long)(uintptr_t)cl + (unsigned long long)(lane * 16u);
            ASYNC_CP(0);    ASYNC_CP(512);  ASYNC_CP(1024); ASYNC_CP(1536);
            ASYNC_CP(2048); ASYNC_CP(2560); ASYNC_CP(3072); ASYNC_CP(3584);
            ASYNC_CP(4096); ASYNC_CP(4608); ASYNC_CP(5120); ASYNC_CP(5632);
            ASYNC_CP(6144); ASYNC_CP(6656); ASYNC_CP(7168); ASYNC_CP(7680);
        }

        // hv^T as B-fragments (K=channel, N=position): lane holds pos p,
        // channels {2h, 2h+1} (frag0) and {4+2h, 5+2h} (frag1)
        v2f hb0, hb1;
        hb0.x = zl[(2 * h + 0) * WH_];
        hb0.y = zl[(2 * h + 1) * WH_];
        hb1.x = zl[(4 + 2 * h + 0) * WH_];
        hb1.y = zl[(4 + 2 * h + 1) * WH_];
        float hn_part = hb0.x * hb0.x + hb0.y * hb0.y + hb1.x * hb1.x + hb1.y * hb1.y;
        const float hn2 = hn_part + xor16f(hn_part);   // ||hv[p]||^2

        // drain the async staging before any LDS read
        asm volatile("s_wait_asynccnt 0x0" ::: "memory");

        v8f   acc   = {0.f, 0.f, 0.f, 0.f, 0.f, 0.f, 0.f, 0.f}; // numerator (D: N=pos, M=ch)
        float denom = 0.f;
        float bestd = 3.402823e38f;
        int   besti = 0;

        for (int nc = 0; nc < 16; ++nc) {
            const int cb16 = nc << 4;

            // codes chunk as A-fragments (M=code, K=channel), from LDS
            const float* cp = clds + (cb16 + p) * CD_;
            v2f ca0 = *(const v2f*)(cp + 2 * h);
            v2f ca1 = *(const v2f*)(cp + 4 + 2 * h);
            float cn_part = ca0.x * ca0.x + ca0.y * ca0.y + ca1.x * ca1.x + ca1.y * ca1.y;
            float cn2 = cn_part + xor16f(cn_part);     // ||code[cb16+p]||^2

            // S^T = codes_chunk(16x8) * hv^T(8x16): two chained K=4 f32 WMMAs
            v8f d = {0.f, 0.f, 0.f, 0.f, 0.f, 0.f, 0.f, 0.f};
            d = __builtin_amdgcn_wmma_f32_16x16x4_f32(false, ca0, false, hb0,
                                                      (short)0, d, false, false);
            d = __builtin_amdgcn_wmma_f32_16x16x4_f32(false, ca1, false, hb1,
                                                      (short)0, d, false, false);
            // d[v] = <code cb16+v+8h, hv[p]>

            float w[8];
            #pragma unroll
            for (int v = 0; v < 8; ++v) {
                float cnv  = __shfl(cn2, v + 8 * h, 32);       // cn2 of code cb16+v+8h
                float d2   = hn2 - 2.f * d[v] + cnv;
                // native v_sqrt_f32: args are in [0, ~1e3], far from denormal range,
                // so skip the IEEE range-fixup expansion
                float dist = __builtin_amdgcn_sqrtf(fmaxf(d2, 0.f));
                int   ci   = cb16 + v + 8 * h;
                if (dist < bestd) { bestd = dist; besti = ci; }
                float e = __expf(-dist);                       // dist>=0 -> e<=1, safe
                w[v]  = e;
                denom += e;
            }

            // acc += probs_chunk * codes_chunk : 4 WMMAs, K = 4 codes each
            #pragma unroll
            for (int j = 0; j < 4; ++j) {
                const int cb   = 4 * j;
                const int base = cb & 7;

                // A: codes^T fragment (M=channel<8, K=code cb16+cb+v+2h)
                // branch-free: clamp row index, select 0 for M>=8 lanes
                v2f at;
                {
                    const int    k0 = cb16 + cb + 2 * h;
                    const float* q  = clds + k0 * CD_ + (p & 7);
                    float a0 = q[0];
                    float a1 = q[CD_];
                    at.x = (p < 8) ? a0 : 0.f;
                    at.y = (p < 8) ? a1 : 0.f;
                }
                // B: exp-weights re-fragmented from D layout to B layout.
                // Needed value: code cb+vv+2h at pos p; it lives in half (cb>>3),
                // vgpr (cb&7)+vv+2h of the w[] array.
                float e00 = w[base + 0], e01 = w[base + 2];
                float e10 = w[base + 1], e11 = w[base + 3];
                float t0 = h ? e01 : e00;
                float t1 = h ? e11 : e10;
                float x0 = xor16f(t0);
                float x1 = xor16f(t1);
                const bool same = ((cb >> 3) == h);
                v2f bt;
                bt.x = same ? t0 : x0;
                bt.y = same ? t1 : x1;

                acc = __builtin_amdgcn_wmma_f32_16x16x4_f32(false, at, false, bt,
                                                            (short)0, acc, false, false);
            }
        }

        // cross-half finalize: each half covered codes {0..7}+16c vs {8..15}+16c
        denom += xor16f(denom);
        {
            float od = xor16f(bestd);
            int   oi = __shfl_xor(besti, 16, 32);
            if (od < bestd || (od == bestd && oi < besti)) { bestd = od; besti = oi; }
        }

        const int pos = q0 + p;
        if (h == 0) {
            // acc[v] = numerator[pos p][channel v] (M = v + 8*0)
            const float inv = __builtin_amdgcn_rcpf(denom);    // native v_rcp_f32
            float* so = soft_out + (size_t)pos * C_ + l * CD_;
            v4f s0 = {acc[0] * inv, acc[1] * inv, acc[2] * inv, acc[3] * inv};
            v4f s1 = {acc[4] * inv, acc[5] * inv, acc[6] * inv, acc[7] * inv};
            *(v4f*)(so + 0) = s0;
            *(v4f*)(so + 4) = s1;
        } else {
            // hard output: gather argmin code row from LDS
            const float* cr = clds + besti * CD_;
            float* ho = hard_out + (size_t)pos * C_ + l * CD_;
            *(v4f*)(ho + 0) = *(const v4f*)(cr + 0);
            *(v4f*)(ho + 4) = *(const v4f*)(cr + 4);
            idx_out[(size_t)pos * L_ + l] = besti;
        }
    }
}

extern "C" void kernel_launch(void* const* d_in, const int* in_sizes, int n_in,
                              void* d_out, int out_size, void* d_ws, size_t ws_size,
                              hipStream_t stream)
{
    (void)in_sizes; (void)n_in; (void)out_size; (void)d_ws; (void)ws_size;

    const float* z     = (const float*)d_in[0];   // (8,128,64,64) f32
    const float* codes = (const float*)d_in[1];   // (16,256,8)   f32

    const size_t n_soft = (size_t)B_ * WH_ * C_;  // 4,194,304
    float* soft = (float*)d_out;
    float* hard = soft + n_soft;
    int*   idx  = (int*)(hard + n_soft);          // 524,288 int32

    const int n_tiles = (B_ * WH_) / 16;          // 2048 position tiles
    vq_soft_hard_kernel<<<n_tiles, 128, 0, stream>>>(z, codes, soft, hard, idx);
}